// VQ_73529840107860
// MI455X (gfx1250) — compile-verified
//
#include <hip/hip_runtime.h>
#include <hip/hip_bf16.h>

#define NUM_EMB   1024
#define DIM       256
#define NTOK      65536
#define ROWS_PER_BLOCK 128   // 8 waves * 16 rows
#define CT_STRIDE 260        // padded LDS row stride (floats): conflict-free + 16B aligned
#define NTILES    (NUM_EMB / 16)

typedef __attribute__((ext_vector_type(2))) float v2f;
typedef __attribute__((ext_vector_type(4))) float v4f;
typedef __attribute__((ext_vector_type(8))) float v8f;

// ---------------------------------------------------------------------------
// Kernel 0: per-code squared norms  ||c_k||^2  -> d_ws[0..1023]
// ---------------------------------------------------------------------------
__global__ void vq_cnorm_kernel(const float* __restrict__ codebook,
                                float* __restrict__ cnorm) {
    const int lane = threadIdx.x & 31;
    const int wv   = threadIdx.x >> 5;
    const int code = blockIdx.x * 8 + wv;
    const float* row = codebook + code * DIM;
    float s = 0.0f;
    #pragma unroll
    for (int d = lane; d < DIM; d += 32) {
        float v = row[d];
        s = fmaf(v, v, s);
    }
    #pragma unroll
    for (int m = 1; m < 32; m <<= 1) s += __shfl_xor(s, m, 32);
    if (lane == 0) cnorm[code] = s;
}

// ---------------------------------------------------------------------------
// Kernel 1: zero the loss accumulator (d_out[NTOK*DIM])
// ---------------------------------------------------------------------------
__global__ void vq_init_kernel(float* __restrict__ out) {
    out[NTOK * DIM] = 0.0f;
}

// ---------------------------------------------------------------------------
// Kernel 2: main VQ kernel
//   - A fragments (16x256 f32 per wave) register-resident, read from HBM once
//   - codebook tiles double-buffered through LDS (1 barrier/tile), next tile
//     prefetched into registers before the WMMA chain
//   - sched_group_barrier pipeline: DS reads stay one 4-chunk group ahead of
//     the WMMAs that consume them (LDS latency hidden under XDL issues)
// ---------------------------------------------------------------------------
__global__ void __launch_bounds__(256, 1)
vq_main_kernel(const float* __restrict__ z_e,
               const float* __restrict__ codebook,
               const float* __restrict__ cnorm,
               float* __restrict__ out) {
    __shared__ float ctile[2][16 * CT_STRIDE];   // double-buffered 16x256 tile
    __shared__ int   nearest_sh[8][16];
    __shared__ float loss_sh[8];

    const int tid  = threadIdx.x;
    const int lane = tid & 31;
    const int wv   = tid >> 5;
    const int r    = lane & 15;        // row (A) / column (B) within 16-tile
    const int hk   = (lane >> 4) * 2;  // k sub-offset {0,2} within K=4 chunk

    const int row_g = blockIdx.x * ROWS_PER_BLOCK + wv * 16 + r;
    const float* zrow = z_e + row_g * DIM;

    // ---- preload A fragments: a[c] = (A[r][4c+hk], A[r][4c+hk+1]) ----
    v2f a[64];
    #pragma unroll
    for (int c = 0; c < 64; ++c)
        a[c] = *(const v2f*)(zrow + 4 * c + hk);

    float best_val[8];
    int   best_idx[8];
    #pragma unroll
    for (int v = 0; v < 8; ++v) { best_val[v] = 3.0e38f; best_idx[v] = 0; }

    // staging geometry: thread t owns 16 floats of code row (t>>4)
    const int ro = tid >> 4;
    const int co = (tid & 15) * 16;
    const float* csrc = codebook + ro * DIM + co;        // tile 0 source
    float* cdst[2] = { ctile[0] + ro * CT_STRIDE + co,
                       ctile[1] + ro * CT_STRIDE + co };

    // ---- prologue: stage tile 0 into buffer 0 ----
    {
        v4f t0 = *(const v4f*)(csrc + 0);
        v4f t1 = *(const v4f*)(csrc + 4);
        v4f t2 = *(const v4f*)(csrc + 8);
        v4f t3 = *(const v4f*)(csrc + 12);
        *(v4f*)(cdst[0] + 0)  = t0;
        *(v4f*)(cdst[0] + 4)  = t1;
        *(v4f*)(cdst[0] + 8)  = t2;
        *(v4f*)(cdst[0] + 12) = t3;
    }
    __syncthreads();

    // ---- sweep 64 codebook column tiles, double-buffered ----
    for (int jt = 0; jt < NTILES; ++jt) {
        // prefetch next tile into registers (overlaps the WMMA chain)
        v4f p0, p1, p2, p3;
        if (jt < NTILES - 1) {
            const float* src = csrc + (jt + 1) * 16 * DIM;
            p0 = *(const v4f*)(src + 0);
            p1 = *(const v4f*)(src + 4);
            p2 = *(const v4f*)(src + 8);
            p3 = *(const v4f*)(src + 12);
        }
        const float cn  = cnorm[jt * 16 + r];
        const int   idx = jt * 16 + r;

        // ---- 16x16 tile of x.c^T via 64 chained f32 WMMAs ----
        v8f acc = {};
        const float* bbase = ctile[jt & 1] + r * CT_STRIDE + hk;
        #pragma unroll
        for (int c = 0; c < 64; ++c) {
            v2f b = *(const v2f*)(bbase + 4 * c);
            acc = __builtin_amdgcn_wmma_f32_16x16x4_f32(
                false, a[c], false, b, (short)0, acc, false, false);
        }

        // Pipeline shape for this scheduling region (masks: 0x100 = DS read,
        // 0x008 = MFMA/WMMA). DS reads run one 4-chunk group ahead:
        //   DS(2), 15 x [ DS(2), WMMA(4) ], WMMA(4)
        __builtin_amdgcn_sched_group_barrier(0x100, 2, 0);
        #pragma unroll
        for (int g = 0; g < 15; ++g) {
            __builtin_amdgcn_sched_group_barrier(0x100, 2, 0);
            __builtin_amdgcn_sched_group_barrier(0x008, 4, 0);
        }
        __builtin_amdgcn_sched_group_barrier(0x008, 4, 0);

        // dist = ||c||^2 - 2 x.c   (x-norm constant per row -> dropped)
        #pragma unroll
        for (int v = 0; v < 8; ++v) {
            float dist = fmaf(-2.0f, acc[v], cn);
            if (dist < best_val[v]) { best_val[v] = dist; best_idx[v] = idx; }
        }

        // drain prefetched tile into the other buffer
        if (jt < NTILES - 1) {
            float* dst = cdst[(jt + 1) & 1];
            *(v4f*)(dst + 0)  = p0;
            *(v4f*)(dst + 4)  = p1;
            *(v4f*)(dst + 8)  = p2;
            *(v4f*)(dst + 12) = p3;
        }
        __syncthreads();   // next buffer written AND current buffer consumed
    }

    // ---- per-row argmin across the 16 lanes of each half-wave ----
    #pragma unroll
    for (int v = 0; v < 8; ++v) {
        float bv = best_val[v];
        int   bi = best_idx[v];
        #pragma unroll
        for (int m = 1; m < 16; m <<= 1) {
            float ov = __shfl_xor(bv, m, 32);
            int   oi = __shfl_xor(bi, m, 32);
            if (ov < bv || (ov == bv && oi < bi)) { bv = ov; bi = oi; }
        }
        // lanes 0..15 hold rows 0..7 (v=row), lanes 16..31 hold rows 8..15
        if ((lane & 15) == 0) nearest_sh[wv][(lane >> 4) * 8 + v] = bi;
    }
    __syncthreads();

    // ---- phase 2: gather z_q, straight-through output, loss ----
    const int code = nearest_sh[wv][r];
    const float* qrow = codebook + code * DIM;
    float* orow = out + row_g * DIM;
    float lsum = 0.0f;
    #pragma unroll
    for (int c = 0; c < 64; ++c) {
        v2f q = *(const v2f*)(qrow + 4 * c + hk);
        v2f d = q - a[c];             // stop_grad(z_q - z_e)
        v2f o = a[c] + d;             // z_e + (z_q - z_e)  (matches reference STE)
        *(v2f*)(orow + 4 * c + hk) = o;
        v2f e = a[c] - q;             // (z_e - z_q)
        lsum = fmaf(e.x, e.x, lsum);
        lsum = fmaf(e.y, e.y, lsum);
    }
    #pragma unroll
    for (int m = 1; m < 32; m <<= 1) lsum += __shfl_xor(lsum, m, 32);
    if (lane == 0) loss_sh[wv] = lsum;
    __syncthreads();
    if (tid == 0) {
        float t = 0.0f;
        #pragma unroll
        for (int w = 0; w < 8; ++w) t += loss_sh[w];
        atomicAdd(out + NTOK * DIM, 2.0f * t);   // embedding + commitment
    }
}

// ---------------------------------------------------------------------------
extern "C" void kernel_launch(void* const* d_in, const int* in_sizes, int n_in,
                              void* d_out, int out_size, void* d_ws, size_t ws_size,
                              hipStream_t stream) {
    const float* z_e      = (const float*)d_in[0];
    const float* codebook = (const float*)d_in[1];
    float* out   = (float*)d_out;
    float* cnorm = (float*)d_ws;   // 1024 floats of scratch

    vq_cnorm_kernel<<<NUM_EMB / 8, 256, 0, stream>>>(codebook, cnorm);
    vq_init_kernel<<<1, 1, 0, stream>>>(out);
    vq_main_kernel<<<NTOK / ROWS_PER_BLOCK, 256, 0, stream>>>(z_e, codebook, cnorm, out);
}